// TriPlaneEmbedder_28286654612024
// MI455X (gfx1250) — compile-verified
//
#include <hip/hip_runtime.h>
#include <climits>
#include <math.h>

#define FDIM 64

typedef __attribute__((ext_vector_type(4))) float f32x4;

// ---------------------------------------------------------------------------
// Plane pointer bundle (kernel argument)
// ---------------------------------------------------------------------------
struct Planes {
    const float* xy0; const float* xy1; const float* xy2;
    const float* xz0; const float* xz1; const float* xz2;
    const float* yz0; const float* yz1; const float* yz2;
};

// ---------------------------------------------------------------------------
// break1 / break2 = first index with d > 1.5 (resp. 2.0); INT_MAX sentinel.
// ---------------------------------------------------------------------------
__global__ void init_breaks_kernel(int* __restrict__ breaks) {
    breaks[0] = INT_MAX;
    breaks[1] = INT_MAX;
}

__global__ __launch_bounds__(1024) void breaks_kernel(
    const float* __restrict__ x, const float* __restrict__ rays_o,
    const float* __restrict__ scale, int* __restrict__ breaks, int n)
{
    __shared__ int s1, s2;
    if (threadIdx.x == 0) { s1 = INT_MAX; s2 = INT_MAX; }
    __syncthreads();

    int i = blockIdx.x * blockDim.x + threadIdx.x;
    bool p1 = false, p2 = false;
    if (i < n) {
        // Exactly mirror: d = || rays_o - x*scale ||, compare d>1.5 / d>2.0
        // via d^2 > 2.25 / 4.0 (exact: sqrt monotone, thresholds representable).
        float px = __fmul_rn(x[3 * i + 0], scale[0]);
        float py = __fmul_rn(x[3 * i + 1], scale[1]);
        float pz = __fmul_rn(x[3 * i + 2], scale[2]);
        float dx = __fsub_rn(rays_o[0], px);
        float dy = __fsub_rn(rays_o[1], py);
        float dz = __fsub_rn(rays_o[2], pz);
        float d2 = __fadd_rn(__fadd_rn(__fmul_rn(dx, dx), __fmul_rn(dy, dy)),
                             __fmul_rn(dz, dz));
        p1 = d2 > 2.25f;
        p2 = d2 > 4.0f;
    }
    // wave32 ballot filter: only the lowest predicated lane per wave hits LDS.
    unsigned long long m1 = __ballot(p1);
    unsigned long long m2 = __ballot(p2);
    int lane = threadIdx.x & (warpSize - 1);
    unsigned long long below = (lane == 0) ? 0ull : (~0ull >> (64 - lane));
    if (p1 && !(m1 & below)) atomicMin(&s1, i);
    if (p2 && !(m2 & below)) atomicMin(&s2, i);
    __syncthreads();
    if (threadIdx.x == 0) {
        if (s1 != INT_MAX) atomicMin(&breaks[0], s1);
        if (s2 != INT_MAX) atomicMin(&breaks[1], s2);
    }
}

// ---------------------------------------------------------------------------
// [C,H,W] -> [H,W,C] transpose. Block = 256 threads, tile = 32(W) x 64(C).
// Reads: NT-hinted, coalesced 128B per wave. Writes: fully contiguous 8KB.
// ---------------------------------------------------------------------------
__global__ __launch_bounds__(256) void transpose_chw_hwc(
    const float* __restrict__ in, float* __restrict__ out, int S)
{
    __shared__ float tile[32][FDIM + 1];
    const int x0 = blockIdx.x * 32;
    const int y  = blockIdx.y;
    const int t  = threadIdx.x;
    const int tx = t & 31;        // position along W
    const int c0 = (t >> 5) * 8;  // channel base for this wave
    const size_t HW = (size_t)S * S;
    const float* src = in + (size_t)y * S + x0 + tx;
#pragma unroll
    for (int k = 0; k < 8; ++k) {
        int c = c0 + k;
        tile[tx][c] = __builtin_nontemporal_load(src + (size_t)c * HW);
    }
    __syncthreads();
    float* dst = out + ((size_t)y * S + x0) * FDIM;
#pragma unroll
    for (int k = 0; k < 8; ++k) {
        int j = k * 256 + t;                 // 0..2047, contiguous
        dst[j] = tile[j >> 6][j & 63];
    }
}

// ---------------------------------------------------------------------------
// Bilinear sampling helpers
// ---------------------------------------------------------------------------
struct Corner {
    size_t i00, i01, i10, i11;   // texel index y*S+x (clipped)
    float  w00, w01, w10, w11;   // bilinear weights
    bool   v00, v01, v10, v11;   // validity (zero padding)
};

__device__ __forceinline__ Corner make_corner(float cx, float cy, int S) {
    const float fs = (float)(S - 1);
    // fx = ((cx + 1) * 0.5) * (S-1), same association as reference
    float fx = __fmul_rn(__fmul_rn(__fadd_rn(cx, 1.0f), 0.5f), fs);
    float fy = __fmul_rn(__fmul_rn(__fadd_rn(cy, 1.0f), 0.5f), fs);
    float x0f = floorf(fx), y0f = floorf(fy);
    float wx = __fsub_rn(fx, x0f), wy = __fsub_rn(fy, y0f);
    int x0 = (int)x0f, y0 = (int)y0f;
    int x1 = x0 + 1, y1 = y0 + 1;
    bool vx0 = ((unsigned)x0 < (unsigned)S);
    bool vx1 = ((unsigned)x1 < (unsigned)S);
    bool vy0 = ((unsigned)y0 < (unsigned)S);
    bool vy1 = ((unsigned)y1 < (unsigned)S);
    int cx0 = min(max(x0, 0), S - 1), cx1 = min(max(x1, 0), S - 1);
    int cy0 = min(max(y0, 0), S - 1), cy1 = min(max(y1, 0), S - 1);
    Corner k;
    size_t r0 = (size_t)cy0 * S, r1 = (size_t)cy1 * S;
    k.i00 = r0 + cx0; k.i01 = r0 + cx1; k.i10 = r1 + cx0; k.i11 = r1 + cx1;
    float omx = 1.0f - wx, omy = 1.0f - wy;
    k.w00 = omx * omy; k.w01 = wx * omy; k.w10 = omx * wy; k.w11 = wx * wy;
    k.v00 = vx0 && vy0; k.v01 = vx1 && vy0; k.v10 = vx0 && vy1; k.v11 = vx1 && vy1;
    return k;
}

__device__ __forceinline__ f32x4 f4zero() { return (f32x4){0.f, 0.f, 0.f, 0.f}; }

// HWC path: one texel = 64 contiguous floats; lane q grabs its float4 slice.
__device__ __forceinline__ f32x4 samp_hwc(const float* __restrict__ p, int S,
                                          const Corner& k, int q) {
    (void)S;
    f32x4 v00 = k.v00 ? ((const f32x4*)(p + k.i00 * FDIM))[q] : f4zero();
    f32x4 v01 = k.v01 ? ((const f32x4*)(p + k.i01 * FDIM))[q] : f4zero();
    f32x4 v10 = k.v10 ? ((const f32x4*)(p + k.i10 * FDIM))[q] : f4zero();
    f32x4 v11 = k.v11 ? ((const f32x4*)(p + k.i11 * FDIM))[q] : f4zero();
    f32x4 r;
#pragma unroll
    for (int j = 0; j < 4; ++j) {
        r[j] = fmaf(v11[j], k.w11,
               fmaf(v10[j], k.w10,
               fmaf(v01[j], k.w01, v00[j] * k.w00)));
    }
    return r;
}

// CHW fallback path: channels strided by H*W (used only if ws too small).
__device__ __forceinline__ f32x4 samp_chw(const float* __restrict__ p, int S,
                                          const Corner& k, int q) {
    const size_t HW = (size_t)S * S;
    const float* pc = p + (size_t)(q * 4) * HW;
    f32x4 r;
#pragma unroll
    for (int j = 0; j < 4; ++j) {
        const float* pj = pc + (size_t)j * HW;
        float v00 = k.v00 ? pj[k.i00] : 0.f;
        float v01 = k.v01 ? pj[k.i01] : 0.f;
        float v10 = k.v10 ? pj[k.i10] : 0.f;
        float v11 = k.v11 ? pj[k.i11] : 0.f;
        r[j] = fmaf(v11, k.w11, fmaf(v10, k.w10, fmaf(v01, k.w01, v00 * k.w00)));
    }
    return r;
}

template <bool HWC>
__device__ __forceinline__ f32x4 tri(const float* __restrict__ p0,
                                     const float* __restrict__ p1,
                                     const float* __restrict__ p2,
                                     float cx, float cy, int level, int q, int S0) {
    Corner k0 = make_corner(cx, cy, S0);
    f32x4 s = HWC ? samp_hwc(p0, S0, k0, q) : samp_chw(p0, S0, k0, q);
    if (level >= 1) {
        Corner k1 = make_corner(cx, cy, S0 >> 1);
        s = s + (HWC ? samp_hwc(p1, S0 >> 1, k1, q) : samp_chw(p1, S0 >> 1, k1, q));
    }
    if (level >= 2) {
        Corner k2 = make_corner(cx, cy, S0 >> 2);
        s = s + (HWC ? samp_hwc(p2, S0 >> 2, k2, q) : samp_chw(p2, S0 >> 2, k2, q));
    }
    return s;
}

// ---------------------------------------------------------------------------
// Main sampler: 16 lanes per point, 4 channels per lane (float4).
// ---------------------------------------------------------------------------
template <bool HWC>
__global__ __launch_bounds__(256) void triplane_sample(
    const float* __restrict__ x, const float* __restrict__ scale,
    const int* __restrict__ breaks, Planes P,
    float* __restrict__ out, int n, int S0)
{
    int t = blockIdx.x * 256 + threadIdx.x;
    int i = t >> 4;       // point index
    int q = t & 15;       // channel quad (4 channels)
    if (i >= n) return;

    // warm L2 for upcoming point coordinates (gfx1250 global_prefetch path)
    if ((threadIdx.x & 63) == 0)
        __builtin_prefetch(x + 3 * (size_t)(i + 4096), 0, 3);

    int b1 = breaks[0]; if (b1 == INT_MAX) b1 = 0;   // c1.any() false -> 0
    int b2 = breaks[1]; if (b2 == INT_MAX) b2 = 0;
    int level = (i < b1) ? 0 : ((i < b2) ? 1 : 2);

    float sx = __fmul_rn(x[3 * i + 0], scale[0]);
    float sy = __fmul_rn(x[3 * i + 1], scale[1]);
    float sz = __fmul_rn(x[3 * i + 2], scale[2]);

    f32x4 txy = tri<HWC>(P.xy0, P.xy1, P.xy2, sx, sy, level, q, S0);
    f32x4 txz = tri<HWC>(P.xz0, P.xz1, P.xz2, sx, sz, level, q, S0);
    f32x4 tyz = tri<HWC>(P.yz0, P.yz1, P.yz2, sy, sz, level, q, S0);
    f32x4 res = (txy + txz) + tyz;   // same association as reference

    // write-once 64MB output: keep it out of L2 (NT store, global_store_b128)
    __builtin_nontemporal_store(res, (f32x4*)out + (size_t)i * 16 + q);
}

// ---------------------------------------------------------------------------
// Host launcher
// ---------------------------------------------------------------------------
extern "C" void kernel_launch(void* const* d_in, const int* in_sizes, int n_in,
                              void* d_out, int out_size, void* d_ws, size_t ws_size,
                              hipStream_t stream) {
    (void)n_in; (void)out_size;
    const float* x     = (const float*)d_in[0];
    const float* rays  = (const float*)d_in[1];
    const float* scale = (const float*)d_in[2];
    const float* xy0 = (const float*)d_in[3];
    const float* yz0 = (const float*)d_in[4];
    const float* xz0 = (const float*)d_in[5];
    const float* xy1 = (const float*)d_in[6];
    const float* yz1 = (const float*)d_in[7];
    const float* xz1 = (const float*)d_in[8];
    const float* xy2 = (const float*)d_in[9];
    const float* yz2 = (const float*)d_in[10];
    const float* xz2 = (const float*)d_in[11];

    const int n  = in_sizes[0] / 3;
    const int S0 = (int)(sqrt((double)(in_sizes[3] / FDIM)) + 0.5);
    const int S1 = S0 >> 1, S2 = S0 >> 2;

    int* breaks = (int*)d_ws;
    init_breaks_kernel<<<1, 1, 0, stream>>>(breaks);
    breaks_kernel<<<(n + 1023) / 1024, 1024, 0, stream>>>(x, rays, scale, breaks, n);

    const size_t sz0 = (size_t)FDIM * S0 * S0;
    const size_t sz1 = (size_t)FDIM * S1 * S1;
    const size_t sz2 = (size_t)FDIM * S2 * S2;
    const size_t need = 256 + 3 * (sz0 + sz1 + sz2) * sizeof(float);
    const bool use_hwc = (ws_size >= need);

    const int total = n * 16;
    const int blocks = (total + 255) / 256;

    Planes P;
    if (use_hwc) {
        float* p = (float*)((char*)d_ws + 256);
        float* t_xy0 = p; p += sz0;
        float* t_xz0 = p; p += sz0;
        float* t_yz0 = p; p += sz0;
        float* t_xy1 = p; p += sz1;
        float* t_xz1 = p; p += sz1;
        float* t_yz1 = p; p += sz1;
        float* t_xy2 = p; p += sz2;
        float* t_xz2 = p; p += sz2;
        float* t_yz2 = p; p += sz2;

        transpose_chw_hwc<<<dim3(S0 / 32, S0), 256, 0, stream>>>(xy0, t_xy0, S0);
        transpose_chw_hwc<<<dim3(S0 / 32, S0), 256, 0, stream>>>(xz0, t_xz0, S0);
        transpose_chw_hwc<<<dim3(S0 / 32, S0), 256, 0, stream>>>(yz0, t_yz0, S0);
        transpose_chw_hwc<<<dim3(S1 / 32, S1), 256, 0, stream>>>(xy1, t_xy1, S1);
        transpose_chw_hwc<<<dim3(S1 / 32, S1), 256, 0, stream>>>(xz1, t_xz1, S1);
        transpose_chw_hwc<<<dim3(S1 / 32, S1), 256, 0, stream>>>(yz1, t_yz1, S1);
        transpose_chw_hwc<<<dim3(S2 / 32, S2), 256, 0, stream>>>(xy2, t_xy2, S2);
        transpose_chw_hwc<<<dim3(S2 / 32, S2), 256, 0, stream>>>(xz2, t_xz2, S2);
        transpose_chw_hwc<<<dim3(S2 / 32, S2), 256, 0, stream>>>(yz2, t_yz2, S2);

        P.xy0 = t_xy0; P.xy1 = t_xy1; P.xy2 = t_xy2;
        P.xz0 = t_xz0; P.xz1 = t_xz1; P.xz2 = t_xz2;
        P.yz0 = t_yz0; P.yz1 = t_yz1; P.yz2 = t_yz2;
        triplane_sample<true><<<blocks, 256, 0, stream>>>(x, scale, breaks, P,
                                                          (float*)d_out, n, S0);
    } else {
        P.xy0 = xy0; P.xy1 = xy1; P.xy2 = xy2;
        P.xz0 = xz0; P.xz1 = xz1; P.xz2 = xz2;
        P.yz0 = yz0; P.yz1 = yz1; P.yz2 = yz2;
        triplane_sample<false><<<blocks, 256, 0, stream>>>(x, scale, breaks, P,
                                                           (float*)d_out, n, S0);
    }
}